// ProposalLayer_44032004719021
// MI455X (gfx1250) — compile-verified
//
#include <hip/hip_runtime.h>
#include <stdint.h>

// ---------------- problem constants ----------------
#define BATCH   8
#define NANCH   261888
#define PRE_NMS 6000
#define PCAP    6016          // PRE_NMS padded to a multiple of 16
#define NTILE   376           // PCAP / 16
#define NPROP   1000

typedef __attribute__((ext_vector_type(2))) float v2f;
typedef __attribute__((ext_vector_type(8))) float v8f;

// ---------------- workspace layout (bytes) ----------------
// state : 8 batches x 8 u32   {prefix, k_rem, thresh, cnt_gt, cnt_eq, need_eq, -, -}
// hist  : 3 levels x 8 x 2048 u32
// boxes : 8 x 6016 float4 (clipped y1,x1,y2,x2)
// scores: 8 x 6016 float
// sidx  : 8 x 6016 u32 (original anchor index, for argmax tie-break)
// areas : 8 x 6016 float
// mask  : 8 x 6016 rows x 376 u16 (suppression bit matrix)
static constexpr size_t WS_STATE  = 0;
static constexpr size_t WS_HIST   = 256;
static constexpr size_t WS_BOXES  = WS_HIST   + (size_t)3*8*2048*4;       // 196864
static constexpr size_t WS_SCORES = WS_BOXES  + (size_t)8*PCAP*16;        // +770048
static constexpr size_t WS_SIDX   = WS_SCORES + (size_t)8*PCAP*4;
static constexpr size_t WS_AREAS  = WS_SIDX   + (size_t)8*PCAP*4;
static constexpr size_t WS_MASK   = WS_AREAS  + (size_t)8*PCAP*4;

// ---------------- gfx1250 async global->LDS helpers ----------------
__device__ __forceinline__ unsigned lds_off(const void* p) {
  // flat shared pointer: low 32 bits are the LDS byte address
  return (unsigned)(unsigned long long)p;
}
__device__ __forceinline__ void async_load_b128(unsigned lds, const void* g) {
  asm volatile("global_load_async_to_lds_b128 %0, %1, off"
               :: "v"(lds), "v"((unsigned long long)g) : "memory");
}
__device__ __forceinline__ void wait_async0() {
  asm volatile("s_wait_asynccnt 0" ::: "memory");
}

// ---------------- init ----------------
__global__ void init_kernel(unsigned* hist, float* scores, unsigned* sidx,
                            float* areas, float4* boxes, unsigned* state) {
  const int gid = blockIdx.x * blockDim.x + threadIdx.x;
  const int stride = gridDim.x * blockDim.x;
  for (int i = gid; i < 3*8*2048; i += stride) hist[i] = 0u;
  for (int i = gid; i < 8*PCAP; i += stride) {
    scores[i] = -1.0f;
    sidx[i]   = 0u;
    areas[i]  = 0.0f;
    boxes[i]  = make_float4(0.f, 0.f, 0.f, 0.f);
  }
  if (gid < 8) {
    unsigned* st = state + gid*8;
    st[0] = 0u;        // prefix
    st[1] = PRE_NMS;   // k remaining
    st[2] = 0u; st[3] = 0u; st[4] = 0u; st[5] = 0u; st[6] = 0u; st[7] = 0u;
  }
}

// ---------------- histogram passes (exact top-k threshold) ----------------
// scores in [0,1] => float bits are monotonic; refine 11 + 11 + 9 = 31 bits.
template <int LVL>
__global__ __launch_bounds__(256) void hist_kernel(const float* probs, unsigned* hist,
                                                   const unsigned* state) {
  const int b = blockIdx.y;
  __shared__ unsigned lh[2048];
  for (int t = threadIdx.x; t < 2048; t += 256) lh[t] = 0u;
  __syncthreads();
  const unsigned pre = state[b*8 + 0];
  const float* pr = probs + (size_t)b * NANCH * 2;
  for (int n = blockIdx.x*256 + threadIdx.x; n < NANCH; n += gridDim.x*256) {
    const unsigned p = __float_as_uint(pr[2*n + 1]);
    if (LVL == 1) {
      atomicAdd(&lh[p >> 20], 1u);
    } else if (LVL == 2) {
      if ((p >> 20) == pre) atomicAdd(&lh[(p >> 9) & 2047u], 1u);
    } else {
      if ((p >> 9) == pre) atomicAdd(&lh[p & 511u], 1u);
    }
  }
  __syncthreads();
  unsigned* gh = hist + (size_t)b * 2048;
  for (int t = threadIdx.x; t < 2048; t += 256)
    if (lh[t]) atomicAdd(&gh[t], lh[t]);
}

template <int LVL>
__global__ void scan_kernel(const unsigned* hist, unsigned* state) {
  const int b = threadIdx.x;
  if (b >= 8) return;
  unsigned* st = state + b*8;
  const unsigned k = st[1];
  const int NB = (LVL == 3) ? 512 : 2048;
  const unsigned* h = hist + (size_t)b * 2048;
  unsigned cum = 0, sel = 0;
  for (int t = NB - 1; t >= 0; --t) {
    const unsigned c = h[t];
    if (cum + c >= k) { sel = (unsigned)t; break; }
    cum += c;
  }
  const unsigned bits = (LVL == 3) ? 9u : 11u;
  st[0] = (st[0] << bits) | sel;
  st[1] = k - cum;                 // how many to take inside selected bucket
  if (LVL == 3) {
    st[2] = st[0];                 // exact bit pattern of 6000th-largest score
    st[3] = 0u;                    // cnt_gt
    st[4] = 0u;                    // cnt_eq
    st[5] = st[1];                 // need_eq
  }
}

// ---------------- compaction: gather top-6000, apply deltas, clip ----------------
__global__ __launch_bounds__(256) void compact_kernel(const float* probs, const float* bbox,
                                                      const float* anch, unsigned* state,
                                                      float4* boxes, float* scores,
                                                      unsigned* sidx, float* areas) {
  const long long total = (long long)BATCH * NANCH;
  const long long stride = (long long)gridDim.x * blockDim.x;
  for (long long i = (long long)blockIdx.x * blockDim.x + threadIdx.x; i < total; i += stride) {
    const int b = (int)(i / NANCH);
    const int n = (int)(i % NANCH);
    const float s = probs[2*i + 1];
    const unsigned p = __float_as_uint(s);
    unsigned* st = state + b*8;
    const unsigned thr = st[2];
    const unsigned need_eq = st[5];
    const unsigned nGT = (unsigned)PRE_NMS - need_eq;
    int slot = -1;
    if (p > thr) {
      const unsigned g = atomicAdd(&st[3], 1u);
      if (g < nGT) slot = (int)g;
    } else if (p == thr) {
      const unsigned e = atomicAdd(&st[4], 1u);
      if (e < need_eq) slot = (int)(nGT + e);
    }
    if (slot >= 0) {
      const float dy = bbox[4*i + 0] * 0.1f;
      const float dx = bbox[4*i + 1] * 0.1f;
      const float dh = bbox[4*i + 2] * 0.2f;
      const float dw = bbox[4*i + 3] * 0.2f;
      const float ay1 = anch[4*i + 0], ax1 = anch[4*i + 1];
      const float ay2 = anch[4*i + 2], ax2 = anch[4*i + 3];
      float h = ay2 - ay1, w = ax2 - ax1;
      const float cy = ay1 + 0.5f*h + dy*h;
      const float cx = ax1 + 0.5f*w + dx*w;
      h *= expf(dh);
      w *= expf(dw);
      float y1 = cy - 0.5f*h, x1 = cx - 0.5f*w;
      float y2 = y1 + h,      x2 = x1 + w;
      y1 = fminf(fmaxf(y1, 0.f), 1.f);
      x1 = fminf(fmaxf(x1, 0.f), 1.f);
      y2 = fminf(fmaxf(y2, 0.f), 1.f);
      x2 = fminf(fmaxf(x2, 0.f), 1.f);
      const size_t o = (size_t)b*PCAP + (size_t)slot;
      boxes[o]  = make_float4(y1, x1, y2, x2);
      scores[o] = s;
      sidx[o]   = (unsigned)n;
      areas[o]  = (y2 - y1) * (x2 - x1);
    }
  }
}

// ---------------- suppression mask: one wave per 16x16 tile, WMMA for area sums ----
// D[i][j] = area_i + area_j  via  A(16x4)=[area_i,1,0,0], B(4x16)=[1;area_j;0;0]
// IoU >= 0.7  <=>  inter >= 0.7*(area_i+area_j - inter + EPS)   (denominator > 0)
__global__ __launch_bounds__(256) void mask_kernel(const float4* boxes, const float* areas,
                                                   unsigned short* mask) {
  const int b    = blockIdx.y;
  const int lane = threadIdx.x & 31;
  const int wave = threadIdx.x >> 5;
  const int tile = blockIdx.x * 8 + wave;        // grid sized so tile < NTILE*NTILE
  const int ti = tile / NTILE, tj = tile % NTILE;
  const int i0 = ti * 16, j0 = tj * 16;
  const float4* bx = boxes + (size_t)b * PCAP;
  const float*  ar = areas + (size_t)b * PCAP;
  const bool lo = lane < 16;

  // A 16x4 f32 layout: lanes 0-15 = M rows, VGPR0/1 = K0/K1; lanes 16-31 = K2/K3 (zero)
  v2f A;  A[0]  = lo ? ar[i0 + lane] : 0.0f;  A[1]  = lo ? 1.0f : 0.0f;
  // B 4x16 f32 layout (mirrored): lanes 0-15 hold rows K0/K1 across N columns
  v2f Bm; Bm[0] = lo ? 1.0f : 0.0f;           Bm[1] = lo ? ar[j0 + lane] : 0.0f;
  v8f C = {0.f,0.f,0.f,0.f,0.f,0.f,0.f,0.f};
  v8f D = __builtin_amdgcn_wmma_f32_16x16x4_f32(false, A, false, Bm,
                                                (short)0, C, false, false);

  const float4 bj = bx[j0 + (lane & 15)];       // this lane's column box
  const int mrow0 = i0 + (lo ? 0 : 8);          // rows this lane's D values belong to
  unsigned short* mb = mask + (size_t)b * PCAP * NTILE;
#pragma unroll
  for (int v = 0; v < 8; ++v) {
    const float4 bi = bx[mrow0 + v];
    const float yy1 = fmaxf(bi.x, bj.x);
    const float xx1 = fmaxf(bi.y, bj.y);
    const float yy2 = fminf(bi.z, bj.z);
    const float xx2 = fminf(bi.w, bj.w);
    const float inter = fmaxf(yy2 - yy1, 0.0f) * fmaxf(xx2 - xx1, 0.0f);
    const bool pred = inter >= 0.7f * (D[v] - inter + 1e-8f);
    const unsigned bal = __builtin_amdgcn_ballot_w32(pred);
    // low 16 bits: row (i0+v); high 16 bits: row (i0+v+8); 16 columns at tile tj
    if (lane == 0)  mb[(size_t)(i0 + v)     * NTILE + tj] = (unsigned short)(bal & 0xFFFFu);
    if (lane == 16) mb[(size_t)(i0 + v + 8) * NTILE + tj] = (unsigned short)(bal >> 16);
  }
}

// ---------------- NMS: one workgroup per batch, scores LDS-resident ----------------
__global__ __launch_bounds__(256) void nms_kernel(const float4* boxes, const float* scores,
                                                  const unsigned* sidx,
                                                  const unsigned short* mask, float4* out) {
  const int b = blockIdx.x;
  const int tid = threadIdx.x;
  __shared__ __align__(16) float    sc[PCAP];
  __shared__ __align__(16) unsigned si[PCAP];
  __shared__ unsigned long long rk[256];
  __shared__ int rp[256];

  // async-stage scores + tie-break indices into LDS (gfx1250 ASYNC path)
  const float*    sg = scores + (size_t)b * PCAP;
  const unsigned* ig = sidx   + (size_t)b * PCAP;
  const unsigned sc0 = lds_off(sc);
  const unsigned si0 = lds_off(si);
  for (int k = tid; k < PCAP/4; k += 256) {
    async_load_b128(sc0 + k*16, sg + k*4);
    async_load_b128(si0 + k*16, ig + k*4);
  }
  wait_async0();
  __syncthreads();

  const unsigned short* mrows = mask + (size_t)b * PCAP * NTILE;
  const float4* bxb = boxes + (size_t)b * PCAP;
  float4* ob = out + (size_t)b * NPROP;

  for (int p = 0; p < NPROP; ++p) {
    // argmax over live scores; tie-break: smallest original anchor index
    unsigned long long bk = 0ull;
    int bp = -1;
    for (int k = tid; k < PCAP; k += 256) {
      const float s = sc[k];
      if (s >= 0.5f) {
        const unsigned long long key =
            ((unsigned long long)__float_as_uint(s) << 32) | (unsigned)(~si[k]);
        if (key > bk) { bk = key; bp = k; }
      }
    }
    rk[tid] = bk; rp[tid] = bp;
    __syncthreads();
    for (int off = 128; off > 0; off >>= 1) {
      if (tid < off && rk[tid + off] > rk[tid]) { rk[tid] = rk[tid + off]; rp[tid] = rp[tid + off]; }
      __syncthreads();
    }
    const int w = rp[0];
    if (w < 0) {                       // no live score left: zero-pad remainder
      for (int q = p + tid; q < NPROP; q += 256) ob[q] = make_float4(0.f, 0.f, 0.f, 0.f);
      break;
    }
    if (tid == 0) { ob[p] = bxb[w]; sc[w] = -1.0f; }
    // suppress everything with IoU >= 0.7 against the pick (precomputed row)
    const unsigned short* mr = mrows + (size_t)w * NTILE;
    for (int h = tid; h < NTILE; h += 256) {
      unsigned bits = mr[h];
      while (bits) {
        const int t = __ffs(bits) - 1;
        sc[h*16 + t] = -1.0f;
        bits &= bits - 1;
      }
    }
    __syncthreads();
  }
}

// ---------------- host entry ----------------
extern "C" void kernel_launch(void* const* d_in, const int* in_sizes, int n_in,
                              void* d_out, int out_size, void* d_ws, size_t ws_size,
                              hipStream_t stream) {
  (void)in_sizes; (void)n_in; (void)out_size; (void)ws_size;
  const float* probs = (const float*)d_in[0];  // [B,N,2]
  const float* bbox  = (const float*)d_in[1];  // [B,N,4]
  const float* anch  = (const float*)d_in[2];  // [B,N,4]
  char* ws = (char*)d_ws;
  unsigned*       state  = (unsigned*)(ws + WS_STATE);
  unsigned*       hist   = (unsigned*)(ws + WS_HIST);
  float4*         boxes  = (float4*)(ws + WS_BOXES);
  float*          scores = (float*)(ws + WS_SCORES);
  unsigned*       sidx   = (unsigned*)(ws + WS_SIDX);
  float*          areas  = (float*)(ws + WS_AREAS);
  unsigned short* mask   = (unsigned short*)(ws + WS_MASK);
  float4* out = (float4*)d_out;                // [B,1000,4]

  init_kernel<<<256, 256, 0, stream>>>(hist, scores, sidx, areas, boxes, state);

  hist_kernel<1><<<dim3(512, BATCH), 256, 0, stream>>>(probs, hist + 0*8*2048, state);
  scan_kernel<1><<<1, 32, 0, stream>>>(hist + 0*8*2048, state);
  hist_kernel<2><<<dim3(512, BATCH), 256, 0, stream>>>(probs, hist + 1*8*2048, state);
  scan_kernel<2><<<1, 32, 0, stream>>>(hist + 1*8*2048, state);
  hist_kernel<3><<<dim3(512, BATCH), 256, 0, stream>>>(probs, hist + 2*8*2048, state);
  scan_kernel<3><<<1, 32, 0, stream>>>(hist + 2*8*2048, state);

  compact_kernel<<<2048, 256, 0, stream>>>(probs, bbox, anch, state,
                                           boxes, scores, sidx, areas);

  mask_kernel<<<dim3((NTILE*NTILE)/8, BATCH), 256, 0, stream>>>(boxes, areas, mask);

  nms_kernel<<<BATCH, 256, 0, stream>>>(boxes, scores, sidx, mask, out);
}